// novaNet_27401891349123
// MI455X (gfx1250) — compile-verified
//
#include <hip/hip_runtime.h>

typedef float v2f __attribute__((ext_vector_type(2)));
typedef float v8f __attribute__((ext_vector_type(8)));

__device__ __forceinline__ float elu_f(float x) { return x > 0.f ? x : (expf(x) - 1.f); }
__device__ __forceinline__ float gelu_f(float x) {
  const float c = 0.7978845608028654f;
  return 0.5f * x * (1.f + tanhf(c * (x + 0.044715f * x * x * x)));
}
__device__ __forceinline__ float sigm_f(float x) { return 1.f / (1.f + expf(-x)); }

enum { GM_BIAS = 1, GM_BIAS_GELU = 2 };

// ---------------------------------------------------------------------------
// Async global->LDS copy (CDNA5 GLOBAL_LOAD_ASYNC_TO_LDS_B128, ASYNCcnt-tracked).
// dst must be 16B aligned, ndw a multiple of 4, src 16B aligned.
// Generic pointers to LDS carry the LDS byte offset in their low 32 bits.
// ---------------------------------------------------------------------------
__device__ __forceinline__ void async_lds_copy_b128(void* dst, const void* src, int ndw) {
  const unsigned base = (unsigned)(size_t)dst;
  const int n4 = ndw >> 2;
  for (int c4 = threadIdx.x; c4 < n4; c4 += (int)blockDim.x) {
    const unsigned ldsa = base + 16u * (unsigned)c4;
    const unsigned voff = 16u * (unsigned)c4;
    asm volatile("global_load_async_to_lds_b128 %0, %1, %2"
                 :: "v"(ldsa), "v"(voff), "s"(src)
                 : "memory");
  }
}
__device__ __forceinline__ void lds_async_wait() {
  asm volatile("s_wait_asynccnt 0x0" ::: "memory");
}

// C(MT*16 x NT*16) = A(MT*16 x K) @ B(K x NT*16), row-major f32 operands in LDS.
// Uses V_WMMA_F32_16X16X4_F32; tile (mt,nt) pairs distributed across waves.
__device__ __forceinline__ void lds_gemm(const float* A, int lda, const float* B, int ldb,
                                         float* C, int ldc, const float* bias,
                                         int MT, int NT, int K, int mode) {
  const int wave = threadIdx.x >> 5;
  const int nw = blockDim.x >> 5;
  const int lane = threadIdx.x & 31;
  const int half = lane >> 4;
  const int l15 = lane & 15;
  for (int tile = wave; tile < MT * NT; tile += nw) {
    const int mt = tile / NT, nt = tile % NT;
    v8f acc = {0.f, 0.f, 0.f, 0.f, 0.f, 0.f, 0.f, 0.f};
    const float* Ar = A + (mt * 16 + l15) * lda;
    const float* Bc = B + nt * 16 + l15;
    for (int k0 = 0; k0 < K; k0 += 4) {
      const int ka = k0 + half * 2;
      v2f av, bv;
      av.x = Ar[ka];
      av.y = Ar[ka + 1];
      bv.x = Bc[ka * ldb];
      bv.y = Bc[(ka + 1) * ldb];
      acc = __builtin_amdgcn_wmma_f32_16x16x4_f32(false, av, false, bv, (short)0, acc,
                                                  false, false);
    }
    const int col = nt * 16 + l15;
    const float bb = bias ? bias[col] : 0.f;
#pragma unroll
    for (int r = 0; r < 8; ++r) {
      const int row = mt * 16 + half * 8 + r;
      float v = acc[r] + bb;
      if (mode == GM_BIAS_GELU) v = gelu_f(v);
      C[row * ldc + col] = v;
    }
  }
}

// ---------------------------------------------------------------------------
// Stage 1: fused conv1+BN1+depthwise+BN2+ELU+pool8 using the algebraic reorder
//   V[c,tau] = sum_h DW[c,h]*X[h,tau];  Z[c,t] = AL[c]*sum_k W1[c>>1,k]*V[c,t-31+k] + BE[c]
// Grid: (5 time tiles of 448, 256 batches). Output P: (B,32,280).
// ---------------------------------------------------------------------------
__global__ __launch_bounds__(256) void novanet_stage1(
    const float* __restrict__ x, const float* __restrict__ w1,
    const float* __restrict__ bn1, const float* __restrict__ dww,
    const float* __restrict__ bn2, float* __restrict__ P) {
  extern __shared__ float sm[];
  float* V  = sm;            // 32*512
  float* W1 = V + 32 * 512;  // 1024
  float* DW = W1 + 1024;     // 704
  float* AL = DW + 704;      // 32
  float* BE = AL + 32;       // 32
  const int tile = blockIdx.x;
  const int b = blockIdx.y;
  const int tid = threadIdx.x;
  const int t0 = tile * 448;

  // async-stage the small weights; overlap with BN folding below
  async_lds_copy_b128(W1, w1, 1024);
  async_lds_copy_b128(DW, dww, 704);
  if (tid < 32) {
    const int c = tid, oc = c >> 1;
    const float s1 = bn1[oc] * rsqrtf(bn1[48 + oc] + 1e-5f);
    const float t1 = bn1[16 + oc] - bn1[32 + oc] * s1;
    const float s2 = bn2[c] * rsqrtf(bn2[96 + c] + 1e-5f);
    const float t2 = bn2[32 + c] - bn2[64 + c] * s2;
    float ds = 0.f;
    for (int h = 0; h < 22; ++h) ds += dww[c * 22 + h];
    AL[c] = s1 * s2;
    BE[c] = t1 * ds * s2 + t2;
  }
  lds_async_wait();
  __syncthreads();

  // Phase A: channel mix 22 -> 32 for 512 halo'd time points
  const float* xb = x + (size_t)b * 22 * 2240;
#pragma unroll
  for (int it = 0; it < 2; ++it) {
    const int tl = tid + it * 256;
    const int tg = t0 - 31 + tl;
    const bool inr = (tg >= 0) && (tg < 2240);
    float xr[22];
#pragma unroll
    for (int h = 0; h < 22; ++h) xr[h] = inr ? xb[h * 2240 + tg] : 0.f;
    for (int c = 0; c < 32; ++c) {
      float acc = 0.f;
#pragma unroll
      for (int h = 0; h < 22; ++h) acc += DW[c * 22 + h] * xr[h];
      V[c * 512 + tl] = acc;
    }
  }
  __syncthreads();

  // Phase B: 64-tap temporal conv + affine + ELU + pool8. 32c * 56tp = 1792 outputs.
#pragma unroll
  for (int it = 0; it < 7; ++it) {
    const int idx = tid + it * 256;
    const int c = idx / 56, tp = idx % 56;
    const float* wf = W1 + (c >> 1) * 64;
    const float* vr = V + c * 512 + tp * 8;
    const float al = AL[c], be = BE[c];
    float sum = 0.f;
#pragma unroll
    for (int j = 0; j < 8; ++j) {
      float acc = 0.f;
#pragma unroll 16
      for (int k = 0; k < 64; ++k) acc += wf[k] * vr[j + k];
      sum += elu_f(al * acc + be);
    }
    P[((size_t)b * 32 + c) * 280 + tile * 56 + tp] = sum * 0.125f;
  }
}

// ---------------------------------------------------------------------------
// Stage 2: conv2 (32->32, k=16) as WMMA GEMM (M=32, K=512, N=288 tiles) + BN3 +
// ELU + pool7 + transpose. One block per batch. Output h: (B,40,32).
// W2 staged with lda=516 (bank-conflict-free A reads); P staged at column
// offset 8 so all async B128 destinations are 16B aligned (B index has +1).
// ---------------------------------------------------------------------------
__global__ __launch_bounds__(256) void novanet_conv2(
    const float* __restrict__ P, const float* __restrict__ w2,
    const float* __restrict__ bn3, float* __restrict__ Hout) {
  extern __shared__ float sm[];
  float* Pp = sm;              // 32*304, Pp[ic][8+t] = P[ic][t], zero pads
  float* W2 = Pp + 32 * 304;   // 32*516
  float* Ob = W2 + 32 * 516;   // 32*288
  const int b = blockIdx.x;
  const int tid = threadIdx.x;

  // async-stage W2 with 512->516 row swizzle (rows stay contiguous in 16B chunks)
  {
    const unsigned base = (unsigned)(size_t)(void*)W2;
    for (int c4 = tid; c4 < 4096; c4 += 256) {
      const int row = c4 >> 7, within = c4 & 127;
      const unsigned ldsa = base + (unsigned)(row * 516 * 4 + within * 16);
      asm volatile("global_load_async_to_lds_b128 %0, %1, %2"
                   :: "v"(ldsa), "v"(16u * (unsigned)c4), "s"(w2) : "memory");
    }
  }
  // async-stage P rows at column byte offset 32 (16B aligned)
  {
    const unsigned base = (unsigned)(size_t)(void*)Pp;
    const float* psrc = P + (size_t)b * 32 * 280;
    for (int c4 = tid; c4 < 2240; c4 += 256) {
      const int ic = c4 / 70, within = c4 % 70;
      const unsigned ldsa = base + (unsigned)(ic * 1216 + 32 + within * 16);
      asm volatile("global_load_async_to_lds_b128 %0, %1, %2"
                   :: "v"(ldsa), "v"(16u * (unsigned)c4), "s"(psrc) : "memory");
    }
  }
  // zero the pad columns (0..7 and 288..303) -- disjoint from async destinations
  for (int idx = tid; idx < 32 * 24; idx += 256) {
    const int ic = idx / 24, j = idx % 24;
    const int col = (j < 8) ? j : (280 + j);
    Pp[ic * 304 + col] = 0.f;
  }
  lds_async_wait();
  __syncthreads();

  const int wave = tid >> 5, lane = tid & 31, half = lane >> 4, l15 = lane & 15;
  for (int tilei = wave; tilei < 36; tilei += 8) {
    const int mt = tilei / 18, nt = tilei % 18;
    v8f acc = {0.f, 0.f, 0.f, 0.f, 0.f, 0.f, 0.f, 0.f};
    const float* Ar = W2 + (mt * 16 + l15) * 516;
    const int tcol = nt * 16 + l15;
    for (int k0 = 0; k0 < 512; k0 += 4) {
      const int ka = k0 + half * 2;
      v2f av, bv;
      av.x = Ar[ka];
      av.y = Ar[ka + 1];
      // B = im2col patch: B[(ic,k)][t] = Ppad[ic][8 + t + k - 7]
      bv.x = Pp[(ka >> 4) * 304 + tcol + (ka & 15) + 1];
      bv.y = Pp[((ka + 1) >> 4) * 304 + tcol + ((ka + 1) & 15) + 1];
      acc = __builtin_amdgcn_wmma_f32_16x16x4_f32(false, av, false, bv, (short)0, acc,
                                                  false, false);
    }
#pragma unroll
    for (int r = 0; r < 8; ++r) Ob[(mt * 16 + half * 8 + r) * 288 + tcol] = acc[r];
  }
  __syncthreads();

  for (int idx = tid; idx < 32 * 40; idx += 256) {
    const int oc = idx / 40, t3 = idx % 40;
    const float s = bn3[oc] * rsqrtf(bn3[96 + oc] + 1e-5f);
    const float sh = bn3[32 + oc] - bn3[64 + oc] * s;
    float sum = 0.f;
#pragma unroll
    for (int j = 0; j < 7; ++j) sum += elu_f(s * Ob[oc * 288 + t3 * 7 + j] + sh);
    Hout[((size_t)b * 40 + t3) * 32 + oc] = sum * (1.f / 7.f);
  }
}

// ---------------------------------------------------------------------------
// Stage 3: 2 Griffin blocks + head, one block per batch. Tokens padded 40->48
// (pad rows kept at zero for every WMMA A-operand buffer). All matmuls via WMMA;
// all weight staging via async global->LDS copies.
// ---------------------------------------------------------------------------
__global__ __launch_bounds__(256) void novanet_blocks(
    const float* __restrict__ Hin, const float* __restrict__ tnorm,
    const float* __restrict__ cnorm, const float* __restrict__ lyw,
    const float* __restrict__ lyb, const float* __restrict__ lxw,
    const float* __restrict__ lxb, const float* __restrict__ cvw,
    const float* __restrict__ cvb, const float* __restrict__ igw,
    const float* __restrict__ igb, const float* __restrict__ agw,
    const float* __restrict__ agb, const float* __restrict__ ap,
    const float* __restrict__ outw, const float* __restrict__ outb,
    const float* __restrict__ upw, const float* __restrict__ upb,
    const float* __restrict__ dnw, const float* __restrict__ dnb,
    const float* __restrict__ hw, const float* __restrict__ hb,
    float* __restrict__ out) {
  extern __shared__ float sm[];
  float* H  = sm;          // 48x32
  float* Tb = H + 1536;    // 48x32
  float* Yb = Tb + 1536;   // 48x32
  float* Ub = Yb + 1536;   // 48x32
  float* Ab = Ub + 1536;   // 48x32
  float* Rb = Ab + 1536;   // 48x32
  float* G0 = Rb + 1536;   // 48x96
  float* G1 = G0 + 4608;   // 48x96
  float* Ws = G1 + 4608;   // 3072 (weight staging / head scratch)
  const int b = blockIdx.x;
  const int tid = threadIdx.x;

  async_lds_copy_b128(H, Hin + (size_t)b * 1280, 1280);
  for (int idx = tid; idx < 256; idx += 256) {  // zero pad rows 40..47
    const int o = 1280 + idx;
    H[o] = 0.f; Tb[o] = 0.f; Yb[o] = 0.f; Ub[o] = 0.f; Ab[o] = 0.f; Rb[o] = 0.f;
  }
  for (int idx = tid; idx < 768; idx += 256) {
    G0[3840 + idx] = 0.f; G1[3840 + idx] = 0.f;
  }
  lds_async_wait();
  __syncthreads();

  for (int i = 0; i < 2; ++i) {
    // rmsnorm(H) -> Tb ; overlap: issue Wy staging first
    async_lds_copy_b128(Ws, lyw + i * 1024, 1024);
    if (tid < 40) {
      float ss = 0.f;
      for (int w = 0; w < 32; ++w) { const float v = H[tid * 32 + w]; ss += v * v; }
      const float r = rsqrtf(ss * (1.f / 32.f) + 1e-6f);
      for (int w = 0; w < 32; ++w)
        Tb[tid * 32 + w] = H[tid * 32 + w] * r * (1.f + tnorm[i * 32 + w]);
    }
    lds_async_wait();
    __syncthreads();
    // y = gelu(Tb @ Wy + by) -> Yb
    lds_gemm(Tb, 32, Ws, 32, Yb, 32, lyb + i * 32, 3, 2, 32, GM_BIAS_GELU);
    __syncthreads();
    // xlin = Tb @ Wx + bx -> Ab
    async_lds_copy_b128(Ws, lxw + i * 1024, 1024);
    lds_async_wait();
    __syncthreads();
    lds_gemm(Tb, 32, Ws, 32, Ab, 32, lxb + i * 32, 3, 2, 32, GM_BIAS);
    __syncthreads();
    // causal conv width 4 -> Ub
    for (int idx = tid; idx < 1280; idx += 256) {
      const int t = idx / 32, c = idx & 31;
      float u = cvb[i * 32 + c];
#pragma unroll
      for (int s = 0; s < 4; ++s) {
        const int ts = t - s;
        if (ts >= 0) u += cvw[(i * 4 + (3 - s)) * 32 + c] * Ab[ts * 32 + c];
      }
      Ub[t * 32 + c] = u;
    }
    __syncthreads();
    // gates (per-head 8x8), xin -> Tb, a -> Ab
    for (int idx = tid; idx < 1280; idx += 256) {
      const int t = idx / 32, c = idx & 31;
      const int h = c >> 3, j = c & 7;
      float gx = igb[i * 32 + h * 8 + j];
      float ga = agb[i * 32 + h * 8 + j];
      for (int m = 0; m < 8; ++m) {
        const float uv = Ub[t * 32 + h * 8 + m];
        gx += uv * igw[((i * 4 + h) * 8 + m) * 8 + j];
        ga += uv * agw[((i * 4 + h) * 8 + m) * 8 + j];
      }
      gx = sigm_f(gx);
      ga = sigm_f(ga);
      const float sp = log1pf(expf(ap[i * 32 + c]));
      const float la = -8.f * ga * sp;
      const float mult = (t == 0) ? 1.f : sqrtf(fmaxf(1.f - expf(2.f * la), 0.f));
      Tb[t * 32 + c] = Ub[t * 32 + c] * gx * mult;
      Ab[t * 32 + c] = expf(la);
    }
    __syncthreads();
    // closed-form scan over T -> Ub (one channel per lane)
    if (tid < 32) {
      float pprod = 1.f, accs = 0.f;
      for (int t = 0; t < 40; ++t) {
        const float ppc = fmaxf(pprod, 1e-7f);
        accs += Tb[t * 32 + tid] / ppc;
        Ub[t * 32 + tid] = ppc * accs;
        pprod *= Ab[t * 32 + tid];
      }
    }
    __syncthreads();
    // prod = u*y -> Ab (re-zero pad rows)
    async_lds_copy_b128(Ws, outw + i * 1024, 1024);
    for (int idx = tid; idx < 1280; idx += 256) Ab[idx] = Ub[idx] * Yb[idx];
    for (int idx = tid; idx < 256; idx += 256) Ab[1280 + idx] = 0.f;
    lds_async_wait();
    __syncthreads();
    // residual = Ab @ Wout + bout + H -> Rb
    lds_gemm(Ab, 32, Ws, 32, Rb, 32, outb + i * 32, 3, 2, 32, GM_BIAS);
    __syncthreads();
    for (int idx = tid; idx < 1280; idx += 256) Rb[idx] += H[idx];
    __syncthreads();
    // cnorm(Rb) -> Tb ; overlap up-weight staging
    async_lds_copy_b128(Ws, upw + i * 6144, 3072);
    if (tid < 40) {
      float ss = 0.f;
      for (int w = 0; w < 32; ++w) { const float v = Rb[tid * 32 + w]; ss += v * v; }
      const float r = rsqrtf(ss * (1.f / 32.f) + 1e-6f);
      for (int w = 0; w < 32; ++w)
        Tb[tid * 32 + w] = Rb[tid * 32 + w] * r * (1.f + cnorm[i * 32 + w]);
    }
    lds_async_wait();
    __syncthreads();
    // MLP up (two 32->96), act = gelu(G0)*G1, down 96->32, + residual -> H
    lds_gemm(Tb, 32, Ws, 96, G0, 96, upb + i * 192, 3, 6, 32, GM_BIAS);
    __syncthreads();
    async_lds_copy_b128(Ws, upw + i * 6144 + 3072, 3072);
    lds_async_wait();
    __syncthreads();
    lds_gemm(Tb, 32, Ws, 96, G1, 96, upb + i * 192 + 96, 3, 6, 32, GM_BIAS);
    __syncthreads();
    async_lds_copy_b128(Ws, dnw + i * 3072, 3072);
    for (int idx = tid; idx < 3840; idx += 256) G0[idx] = gelu_f(G0[idx]) * G1[idx];
    for (int idx = tid; idx < 768; idx += 256) G0[3840 + idx] = 0.f;
    lds_async_wait();
    __syncthreads();
    lds_gemm(G0, 96, Ws, 32, H, 32, dnb + i * 32, 3, 2, 96, GM_BIAS);
    __syncthreads();
    for (int idx = tid; idx < 1280; idx += 256) H[idx] += Rb[idx];
    __syncthreads();
  }

  // head: mean over T, @ head_w + head_b
  float* mb = Ws;
  if (tid < 32) {
    float s = 0.f;
    for (int t = 0; t < 40; ++t) s += H[t * 32 + tid];
    mb[tid] = s * (1.f / 40.f);
  }
  __syncthreads();
  if (tid < 4) {
    float o = hb[tid];
    for (int w = 0; w < 32; ++w) o += mb[w] * hw[w * 4 + tid];
    out[(size_t)b * 4 + tid] = o;
  }
}

extern "C" void kernel_launch(void* const* d_in, const int* in_sizes, int n_in,
                              void* d_out, int out_size, void* d_ws, size_t ws_size,
                              hipStream_t stream) {
  (void)in_sizes; (void)n_in; (void)out_size; (void)ws_size;
  const float* x       = (const float*)d_in[0];
  const float* conv1_w = (const float*)d_in[1];
  const float* bn1     = (const float*)d_in[2];
  const float* dw_w    = (const float*)d_in[3];
  const float* bn2     = (const float*)d_in[4];
  const float* conv2_w = (const float*)d_in[5];
  const float* bn3     = (const float*)d_in[6];
  const float* tnorm   = (const float*)d_in[7];
  const float* cnorm   = (const float*)d_in[8];
  const float* lyw     = (const float*)d_in[9];
  const float* lyb     = (const float*)d_in[10];
  const float* lxw     = (const float*)d_in[11];
  const float* lxb     = (const float*)d_in[12];
  const float* cvw     = (const float*)d_in[13];
  const float* cvb     = (const float*)d_in[14];
  const float* igw     = (const float*)d_in[15];
  const float* igb     = (const float*)d_in[16];
  const float* agw     = (const float*)d_in[17];
  const float* agb     = (const float*)d_in[18];
  const float* ap      = (const float*)d_in[19];
  const float* outw    = (const float*)d_in[20];
  const float* outb    = (const float*)d_in[21];
  const float* upw     = (const float*)d_in[22];
  const float* upb     = (const float*)d_in[23];
  const float* dnw     = (const float*)d_in[24];
  const float* dnb     = (const float*)d_in[25];
  const float* hw      = (const float*)d_in[26];
  const float* hb      = (const float*)d_in[27];

  float* P    = (float*)d_ws;                 // (256,32,280)
  float* Hbuf = P + (size_t)256 * 32 * 280;   // (256,40,32)

  const size_t sh1 = (size_t)(32 * 512 + 1024 + 704 + 64) * sizeof(float);
  const size_t sh2 = (size_t)(32 * 304 + 32 * 516 + 32 * 288) * sizeof(float);
  const size_t sh3 = (size_t)(6 * 1536 + 2 * 4608 + 3072) * sizeof(float);

  novanet_stage1<<<dim3(5, 256), 256, sh1, stream>>>(x, conv1_w, bn1, dw_w, bn2, P);
  novanet_conv2<<<256, 256, sh2, stream>>>(P, conv2_w, bn3, Hbuf);
  novanet_blocks<<<256, 256, sh3, stream>>>(Hbuf, tnorm, cnorm, lyw, lyb, lxw, lxb,
                                            cvw, cvb, igw, igb, agw, agb, ap, outw,
                                            outb, upw, upb, dnw, dnb, hw, hb,
                                            (float*)d_out);
}